// ScatterEdges_15101105013297
// MI455X (gfx1250) — compile-verified
//
#include <hip/hip_runtime.h>
#include <cstdint>

#define D_FEAT   64
#define TPB      256        // 8 wave32 per block
#define E_TILE   64         // edges per LDS tile: 64*64*4 = 16 KB
#define NBLOCKS  4096       // persistent blocks; each runs a 2-deep async pipeline

// ---- gfx1250 async global->LDS path (guarded; host pass sees neither) ----
#if defined(__HIP_DEVICE_COMPILE__) && __has_builtin(__builtin_amdgcn_global_load_async_to_lds_b128)
#define HAVE_ASYNC_LDS 1
#else
#define HAVE_ASYNC_LDS 0
#endif

#if defined(__HIP_DEVICE_COMPILE__) && __has_builtin(__builtin_amdgcn_s_wait_asynccnt)
#define WAIT_ASYNC(N) __builtin_amdgcn_s_wait_asynccnt(N)
#else
#define WAIT_ASYNC(N) asm volatile("s_wait_asynccnt " #N ::: "memory")
#endif

#if HAVE_ASYNC_LDS
typedef int v4i_ty __attribute__((vector_size(16)));
typedef __attribute__((address_space(1))) v4i_ty* gv4_ptr;
typedef __attribute__((address_space(3))) v4i_ty* lv4_ptr;
#endif

__device__ __forceinline__ void atomic_add_f32(float* p, float v) {
  // native global_atomic_add_f32 (no CAS loop); order nondeterminism is the
  // standard tradeoff for scatter-add segment sums.
  unsafeAtomicAdd(p, v);
}

__global__ __launch_bounds__(TPB) void scatter_edges_kernel(
    const float* __restrict__ x,     // [E, 64]
    const float* __restrict__ sw,    // [E]
    const int*   __restrict__ src,   // [E]
    const int*   __restrict__ dst,   // [E]
    float*       __restrict__ out,   // [N, 64]
    int nedges, int ntiles)
{
  const int t = threadIdx.x;

#if HAVE_ASYNC_LDS
  __shared__ float tile[2][E_TILE][D_FEAT];   // 32 KB, double buffered

  // Issue one tile's worth of async DMA: 1024 b128 chunks / 256 threads = 4 per thread
  // (exactly 4 async instructions per wave -> ASYNCcnt += 4).
  auto issue = [&](int ti, int b) {
#pragma unroll
    for (int k = 0; k < 4; ++k) {
      const int  c    = t + k * TPB;                  // b128 chunk id in tile
      long       e    = (long)ti * E_TILE + (c >> 4); // edge this chunk belongs to
      if (e >= nedges) e = nedges - 1;                // clamp: keep instr count fixed
      const float* g = x + e * D_FEAT + (c & 15) * 4;
      const float* l = &tile[b][0][0] + (size_t)c * 4;
      __builtin_amdgcn_global_load_async_to_lds_b128(
          (gv4_ptr)(uintptr_t)g,
          (lv4_ptr)(uint32_t)(uintptr_t)l,
          /*offset=*/0, /*cpol=*/0);
    }
  };

  int buf = 0;
  if ((int)blockIdx.x < ntiles) issue(blockIdx.x, 0);

  for (int ti = blockIdx.x; ti < ntiles; ti += gridDim.x) {
    const int tnext = ti + gridDim.x;
    if (tnext < ntiles) {
      issue(tnext, buf ^ 1);   // kick next tile's DMA first...
      WAIT_ASYNC(4);           // ...then wait: <=4 outstanding => current tile landed
    } else {
      WAIT_ASYNC(0);
    }
    __syncthreads();           // all waves' DMA portions visible

    // Consume: 16 lanes per edge, float4 per lane (ds_load_b128), 4 passes.
    const int colg = (t & 15) * 4;
#pragma unroll
    for (int p = 0; p < 4; ++p) {
      const int  el = (t >> 4) + p * 16;          // edge row within tile
      const long e  = (long)ti * E_TILE + el;
      if (e < nedges) {
        const float s  = sw[e];
        const int   ns = src[e];
        const int   nd = dst[e];
        const float4 v = *(const float4*)&tile[buf][el][colg];
        const float a0 = v.x * s, a1 = v.y * s, a2 = v.z * s, a3 = v.w * s;
        float* po = out + (size_t)ns * D_FEAT + colg;
        float* pq = out + (size_t)nd * D_FEAT + colg;
        atomic_add_f32(po + 0, a0); atomic_add_f32(po + 1, a1);
        atomic_add_f32(po + 2, a2); atomic_add_f32(po + 3, a3);
        atomic_add_f32(pq + 0, a0); atomic_add_f32(pq + 1, a1);
        atomic_add_f32(pq + 2, a2); atomic_add_f32(pq + 3, a3);
      }
    }
    __syncthreads();           // everyone done reading buf before it is refilled
    buf ^= 1;
  }
#else
  // Fallback: direct coalesced float4 loads + native atomics (no LDS staging).
  const long nquads = (long)nedges * (D_FEAT / 4);
  for (long q = (long)blockIdx.x * TPB + t; q < nquads; q += (long)gridDim.x * TPB) {
    const long e    = q >> 4;
    const int  colg = (int)(q & 15) * 4;
    const float s   = sw[e];
    const float4 v  = *(const float4*)(x + e * D_FEAT + colg);
    const float a0 = v.x * s, a1 = v.y * s, a2 = v.z * s, a3 = v.w * s;
    const int ns = src[e], nd = dst[e];
    float* po = out + (size_t)ns * D_FEAT + colg;
    float* pq = out + (size_t)nd * D_FEAT + colg;
    atomic_add_f32(po + 0, a0); atomic_add_f32(po + 1, a1);
    atomic_add_f32(po + 2, a2); atomic_add_f32(po + 3, a3);
    atomic_add_f32(pq + 0, a0); atomic_add_f32(pq + 1, a1);
    atomic_add_f32(pq + 2, a2); atomic_add_f32(pq + 3, a3);
  }
#endif
}

extern "C" void kernel_launch(void* const* d_in, const int* in_sizes, int n_in,
                              void* d_out, int out_size, void* d_ws, size_t ws_size,
                              hipStream_t stream) {
  (void)n_in; (void)d_ws; (void)ws_size;
  const float* x  = (const float*)d_in[0];
  const float* sw = (const float*)d_in[1];
  const int*  esrc = (const int*)d_in[2];
  const int*  edst = (const int*)d_in[3];
  // d_in[4] (species) only supplies N, already encoded in out_size.
  float* out = (float*)d_out;

  const int E      = in_sizes[1];                    // switch has one entry per edge
  const int ntiles = (E + E_TILE - 1) / E_TILE;

  // Atomic accumulation target must start at zero every call (capture-safe).
  (void)hipMemsetAsync(d_out, 0, (size_t)out_size * sizeof(float), stream);

  const int blocks = ntiles < NBLOCKS ? ntiles : NBLOCKS;
  scatter_edges_kernel<<<blocks, TPB, 0, stream>>>(x, sw, esrc, edst, out, E, ntiles);
}